// GCGRUCell2_78400333021777
// MI455X (gfx1250) — compile-verified
//
#include <hip/hip_runtime.h>
#include <hip/hip_bf16.h>
#include <math.h>

// ---------------------------------------------------------------------------
// GCGRU cell (DCRNN diffusion-conv GRU), MI455X / gfx1250, wave32 + WMMA.
//
// Heavy op: einsum('bijs,bjf->bifs') = per (b,s) a (1024x1024)@(1024x66) GEMM,
// done twice (r/u pass, then c pass with state = r*h_prev).
// supports = 268 MB f32, streamed twice -> >=536 MB HBM -> ~23us floor.
// Math (~17.5 GFLOP) mapped to V_WMMA_F32_16X16X4_F32 (native f32, no cvt).
// x0 chunks double-buffered in LDS via GLOBAL_LOAD_ASYNC_TO_LDS (ASYNCcnt);
// A tiles software-pipelined (depth 2) in registers to hide load latency.
// ---------------------------------------------------------------------------

typedef __attribute__((ext_vector_type(2))) float v2f;
typedef __attribute__((ext_vector_type(4))) float v4f;
typedef __attribute__((ext_vector_type(8))) float v8f;

typedef __attribute__((address_space(1))) int as1_int;   // global
typedef __attribute__((address_space(3))) int as3_int;   // LDS

#if defined(__gfx1250__) && __has_builtin(__builtin_amdgcn_global_load_async_to_lds_b32)
#define GC_ASYNC 1
#else
#define GC_ASYNC 0
#endif
#if defined(__gfx1250__) && __has_builtin(__builtin_amdgcn_s_wait_asynccnt)
#define GC_WAITA_BUILTIN 1
#else
#define GC_WAITA_BUILTIN 0
#endif

__device__ __forceinline__ void wait_async0() {
#if GC_ASYNC
#if GC_WAITA_BUILTIN
    __builtin_amdgcn_s_wait_asynccnt(0);
#else
    asm volatile("s_wait_asynccnt 0x0" ::: "memory");
#endif
#endif
}

constexpr int kB  = 32;
constexpr int kN  = 1024;
constexpr int kD  = 2;
constexpr int kU  = 64;
constexpr int kS  = 2;
constexpr int kF  = kD + kU;     // 66
constexpr int kFP = 80;          // padded to 5 f-tiles of 16
constexpr int kJP = 68;          // transposed-LDS j stride: 68 dwords -> the
                                 // 32 b64 lane reads hit 64 distinct banks
constexpr int kROWS = (kD + kU) * (kS + 1);  // 198

// ---------------------------------------------------------------------------
// Diffusion einsum. MODE 0: x0 = [inputs | h_prev] (F=66, NT=5 f-tiles)
//                   MODE 1: x0 = state2 (r*h_prev) (F=64, NT=4 f-tiles)
// Block: 256 threads (8 waves); wave w handles i-rows [i0, i0+16).
// Grid: (N/128, B).  Output xm[b, i, s, 0..NT*16) with row stride fstride.
// ---------------------------------------------------------------------------
template <int NT, int MODE>
__global__ void __launch_bounds__(256)
diffuse_kernel(const float* __restrict__ supports,  // (B,N,N,S)
               const float* __restrict__ inA,       // MODE0: inputs; MODE1: state2
               const float* __restrict__ inB,       // MODE0: h_prev; MODE1: unused
               float* __restrict__ xm,              // (B,N,S,fstride)
               int fstride)
{
    // Transposed x0 chunk: xbufT[buf][f][j - j0], double buffered.
    __shared__ float xbufT[2][kFP][kJP];

    const int b    = blockIdx.y;
    const int tid  = threadIdx.x;
    const int wave = tid >> 5;
    const int lane = tid & 31;
    const int half = lane >> 4;   // 0: lanes 0-15, 1: lanes 16-31
    const int l16  = lane & 15;
    const int i0   = blockIdx.x * 128 + wave * 16;
    const int NFv  = (MODE == 0) ? kF : kU;   // valid feature rows

    // Zero the padding feature rows of both buffers once (never re-written).
    for (int idx = tid; idx < (kFP - NFv) * 64; idx += 256) {
        const int f = NFv + (idx >> 6), jj = idx & 63;
        xbufT[0][f][jj] = 0.0f;
        xbufT[1][f][jj] = 0.0f;
    }

    // Stage x0[j0c : j0c+64, 0:NFv) transposed into buffer `buf`.
    auto stage = [&](int j0c, int buf) {
        for (int idx = tid; idx < NFv * 64; idx += 256) {
            const int f = idx >> 6, jj = idx & 63;
            const int j = j0c + jj;
            const float* src;
            if (MODE == 0)
                src = (f < kD) ? &inA[((size_t)b * kN + j) * kD + f]
                               : &inB[((size_t)b * kN + j) * kU + (f - kD)];
            else
                src = &inA[((size_t)b * kN + j) * kU + f];
#if GC_ASYNC
            __builtin_amdgcn_global_load_async_to_lds_b32(
                (as1_int*)src, (as3_int*)&xbufT[buf][f][jj], 0, 0);
#else
            xbufT[buf][f][jj] = *src;
#endif
        }
    };

    const v8f vz = {0.f, 0.f, 0.f, 0.f, 0.f, 0.f, 0.f, 0.f};
    v8f acc[kS][NT];
#pragma unroll
    for (int s = 0; s < kS; ++s)
#pragma unroll
        for (int t = 0; t < NT; ++t) acc[s][t] = vz;

    // This lane's supports row (A-matrix row M = lane%16 of the i-tile).
    const float* srow =
        supports + ((size_t)b * kN + (i0 + l16)) * (size_t)(kN * kS);

    stage(0, 0);  // prologue: chunk 0 into buffer 0

    for (int c = 0; c < kN / 64; ++c) {
        const int j0  = c * 64;
        const int buf = c & 1;

        wait_async0();       // own async copies for this chunk complete
        __syncthreads();     // all waves' copies complete
        if (c + 1 < kN / 64) stage(j0 + 64, buf ^ 1);  // overlap next stage
        if (j0 + 64 < kN)    // prefetch next chunk of this supports row
            __builtin_prefetch(srow + (size_t)(j0 + 64) * kS, 0, 3);

        // A-tile register pipeline, depth 2: load k+8 while WMMAs consume k.
        // float4 = [(j,s0),(j,s1),(j+1,s0),(j+1,s1)] : deinterleave is this
        // lane's A-slice for BOTH s-tiles (f32 A 16x4 layout: lanes 0-15
        // hold K=0,1; lanes 16-31 hold K=2,3).
        v4f apipe0 = *(const v4f*)(srow + (size_t)(j0 + half * 2) * kS);
        v4f apipe1 = *(const v4f*)(srow + (size_t)(j0 + 4 + half * 2) * kS);
#pragma unroll
        for (int k = 0; k < 64; k += 4) {
            int jn = j0 + k + 8 + half * 2;
            if (jn >= kN) jn = half * 2;  // dummy in-bounds address
            const v4f anext = *(const v4f*)(srow + (size_t)jn * kS);

            const v2f a0 = {apipe0.x, apipe0.z};  // s = 0
            const v2f a1 = {apipe0.y, apipe0.w};  // s = 1
            const int kl = k + half * 2;
#pragma unroll
            for (int t = 0; t < NT; ++t) {
                // B 4x16 f32: lane holds B[half*2, col], B[half*2+1, col];
                // transposed LDS makes this one aligned b64.
                const v2f bt = *(const v2f*)&xbufT[buf][t * 16 + l16][kl];
                acc[0][t] = __builtin_amdgcn_wmma_f32_16x16x4_f32(
                    false, a0, false, bt, (short)0, acc[0][t], false, false);
                acc[1][t] = __builtin_amdgcn_wmma_f32_16x16x4_f32(
                    false, a1, false, bt, (short)0, acc[1][t], false, false);
            }
            apipe0 = apipe1;
            apipe1 = anext;
        }
    }

    // Store. C/D f32 16x16 layout: VGPR r -> M = r + 8*half, N = lane%16.
#pragma unroll
    for (int s = 0; s < kS; ++s)
#pragma unroll
        for (int t = 0; t < NT; ++t)
#pragma unroll
            for (int r = 0; r < 8; ++r) {
                const int i = i0 + r + half * 8;
                xm[(((size_t)b * kN + i) * kS + s) * fstride + t * 16 + l16] =
                    acc[s][t][r];
            }
}

// ---------------------------------------------------------------------------
// Gate pass 1: r,u = sigmoid(x_cat @ K + bias); emit u and state2 = r*h_prev.
// Block: 256 threads = 4 node-rows x 64 output channels.
// x_cat[f*3 + m]: m=0 -> x0, m=1+s -> xm1[...,s] (matches reference reshape).
// ---------------------------------------------------------------------------
__global__ void __launch_bounds__(256)
gates_ru_kernel(const float* __restrict__ inputs, const float* __restrict__ h_prev,
                const float* __restrict__ xm1,
                const float* __restrict__ rk, const float* __restrict__ rb,
                const float* __restrict__ uk, const float* __restrict__ ub,
                float* __restrict__ u_out, float* __restrict__ st2)
{
    __shared__ float xc[4][kROWS];
    const int tid = threadIdx.x;
    const int sub = tid >> 6;
    const int o   = tid & 63;
    const int row = blockIdx.x * 4 + sub;  // = b*N + i

    for (int idx = o; idx < kROWS; idx += 64) {
        const int f = idx / 3, m = idx - 3 * f;
        float v;
        if (m == 0) v = (f < kD) ? inputs[row * kD + f] : h_prev[row * kU + (f - kD)];
        else        v = xm1[((size_t)row * kS + (m - 1)) * kFP + f];
        xc[sub][idx] = v;
    }
    __syncthreads();

    float r = rb[o], u = ub[o];
    for (int k = 0; k < kROWS; ++k) {
        const float x = xc[sub][k];
        r = fmaf(x, rk[k * kU + o], r);
        u = fmaf(x, uk[k * kU + o], u);
    }
    r = 1.0f / (1.0f + __expf(-r));
    u = 1.0f / (1.0f + __expf(-u));
    const float hp = h_prev[row * kU + o];
    st2[row * kU + o]   = r * hp;
    u_out[row * kU + o] = u;
}

// ---------------------------------------------------------------------------
// Gate pass 2: c = tanh(x_cat_c @ ck + cb); h = u*h_prev + (1-u)*c.
// Input channels (f<D) of the diffusion are state-independent -> reuse xm1.
// ---------------------------------------------------------------------------
__global__ void __launch_bounds__(256)
gates_c_kernel(const float* __restrict__ inputs, const float* __restrict__ h_prev,
               const float* __restrict__ xm1, const float* __restrict__ xm2,
               const float* __restrict__ st2, const float* __restrict__ u_ws,
               const float* __restrict__ ck, const float* __restrict__ cb,
               float* __restrict__ out)
{
    __shared__ float xc[4][kROWS];
    const int tid = threadIdx.x;
    const int sub = tid >> 6;
    const int o   = tid & 63;
    const int row = blockIdx.x * 4 + sub;

    for (int idx = o; idx < kROWS; idx += 64) {
        const int f = idx / 3, m = idx - 3 * f;
        float v;
        if (m == 0) {
            v = (f < kD) ? inputs[row * kD + f] : st2[row * kU + (f - kD)];
        } else {
            const int s = m - 1;
            v = (f < kD) ? xm1[((size_t)row * kS + s) * kFP + f]
                         : xm2[((size_t)row * kS + s) * kU + (f - kD)];
        }
        xc[sub][idx] = v;
    }
    __syncthreads();

    float c = cb[o];
    for (int k = 0; k < kROWS; ++k)
        c = fmaf(xc[sub][k], ck[k * kU + o], c);
    c = tanhf(c);

    const float u  = u_ws[row * kU + o];
    const float hp = h_prev[row * kU + o];
    out[row * kU + o] = u * hp + (1.0f - u) * c;
}

// ---------------------------------------------------------------------------
extern "C" void kernel_launch(void* const* d_in, const int* in_sizes, int n_in,
                              void* d_out, int out_size, void* d_ws, size_t ws_size,
                              hipStream_t stream) {
    (void)in_sizes; (void)n_in; (void)out_size; (void)ws_size;
    const float* inputs   = (const float*)d_in[0];  // (B,N,D)
    const float* supports = (const float*)d_in[1];  // (B,N,N,S)
    const float* h_prev   = (const float*)d_in[2];  // (B,N*U)
    const float* rk = (const float*)d_in[3];
    const float* rb = (const float*)d_in[4];
    const float* uk = (const float*)d_in[5];
    const float* ub = (const float*)d_in[6];
    const float* ck = (const float*)d_in[7];
    const float* cb = (const float*)d_in[8];
    float* out = (float*)d_out;                     // (B, N*U)

    // Workspace carve-up (f32): ~55 MB total, fully rewritten each call.
    float* xm1  = (float*)d_ws;                         // B*N*S*kFP
    float* u_ws = xm1  + (size_t)kB * kN * kS * kFP;    // B*N*U
    float* st2  = u_ws + (size_t)kB * kN * kU;          // B*N*U
    float* xm2  = st2  + (size_t)kB * kN * kU;          // B*N*S*kU

    const dim3 gdiff(kN / 128, kB);
    const int  grows = (kB * kN) / 4;

    diffuse_kernel<5, 0><<<gdiff, 256, 0, stream>>>(supports, inputs, h_prev, xm1, kFP);
    gates_ru_kernel<<<grows, 256, 0, stream>>>(inputs, h_prev, xm1, rk, rb, uk, ub,
                                               u_ws, st2);
    diffuse_kernel<4, 1><<<gdiff, 256, 0, stream>>>(supports, st2, nullptr, xm2, kU);
    gates_c_kernel<<<grows, 256, 0, stream>>>(inputs, h_prev, xm1, xm2, st2, u_ws,
                                              ck, cb, out);
}